// tokenTypeLoss_36498632082234
// MI455X (gfx1250) — compile-verified
//
#include <hip/hip_runtime.h>
#include <hip/hip_bf16.h>
#include <stdint.h>

// Problem shape (fixed by the reference)
#define B_DIM 32
#define C_DIM 8192
#define S_DIM 120
#define BS_TOTAL (B_DIM * S_DIM)          // 3840

// k1 tiling
#define NCHUNK 16
#define ROWS_PER_CHUNK (C_DIM / NCHUNK)   // 512
#define WAVES_K1 8
#define THREADS_K1 (WAVES_K1 * 32)        // 256
#define ROWS_PER_WAVE (ROWS_PER_CHUNK / WAVES_K1) // 64
#define TILE_ROWS 8
#define NTILES (ROWS_PER_WAVE / TILE_ROWS)        // 8
#define ROW_BYTES (S_DIM * 4)             // 480
#define TILE_BYTES (TILE_ROWS * ROW_BYTES)        // 3840
#define WAVE_LDS_BYTES (2 * TILE_BYTES)   // 7680 (double buffer)
#define SMEM_FLOATS (WAVES_K1 * WAVE_LDS_BYTES / 4) // 15360 floats = 60 KB

#if defined(__AMDGCN__) && __has_builtin(__builtin_amdgcn_tensor_load_to_lds) && __has_builtin(__builtin_amdgcn_s_wait_tensorcnt)
#define USE_TDM 1
#else
#define USE_TDM 0
#endif

typedef uint32_t tdm_v4u __attribute__((ext_vector_type(4)));
typedef int      tdm_v8i __attribute__((ext_vector_type(8)));
typedef int      tdm_v4i __attribute__((ext_vector_type(4)));

__device__ __forceinline__ float neg_inf_f() { return __int_as_float((int)0xff800000u); }

// Issue one TDM DMA: 2D tile [TILE_ROWS x S_DIM] of f32, row stride S_DIM,
// global -> LDS.  D# packing per CDNA5 ISA §8 (groups 0/1; groups 2/3 zero
// => dims 2..4 unused).  All operands are wave-uniform.
// This toolchain exposes the 6-arg builtin:
//   (uint32x4 g0, int32x8 g1, int32x4 g2, int32x4 g3, int32x8 extra, i32 cpol)
__device__ __forceinline__ void tdm_issue_tile(uint64_t gaddr, uint32_t lds_off) {
#if USE_TDM
  tdm_v4u g0;
  g0.x = 1u;                                   // count=1, user descriptor
  g0.y = lds_off;                              // lds_addr (bytes)
  g0.z = (uint32_t)gaddr;                      // global_addr[31:0]
  g0.w = (uint32_t)((gaddr >> 32) & 0x01FFFFFFu) | 0x80000000u; // addr[56:32] | type=2
  tdm_v8i g1;
  g1[0] = 0x00020000;                          // data_size=2 (4 bytes); no pad/iterate/multicast
  g1[1] = (S_DIM << 16);                       // atomic_barrier_addr=0 | tensor_dim0[15:0]=120
  g1[2] = (TILE_ROWS << 16);                   // tensor_dim0[31:16]=0 | tensor_dim1[15:0]=8
  g1[3] = (S_DIM << 16);                       // tensor_dim1[31:16]=0 | tile_dim0=120
  g1[4] = TILE_ROWS;                           // tile_dim1=8 | tile_dim2=0
  g1[5] = S_DIM;                               // tensor_dim0_stride[31:0]=120
  g1[6] = 0;                                   // stride0[47:32]=0 | stride1[15:0]=0
  g1[7] = 0;                                   // stride1[47:16]=0
  tdm_v4i z4 = {0, 0, 0, 0};
  tdm_v8i z8 = {0, 0, 0, 0, 0, 0, 0, 0};
  __builtin_amdgcn_tensor_load_to_lds(g0, g1, z4, z4, z8, 0);
#else
  (void)gaddr; (void)lds_off;
#endif
}

// ---------------------------------------------------------------------------
// k1: per (b, chunk) block.  Each of 8 waves streams its 64 contiguous rows
// (c-major ascending for argmax tie order) through a double-buffered LDS tile
// via the TDM.  Produces per-(b,chunk,s) partials {colmax, colsum, best, bestc}.
// ---------------------------------------------------------------------------
__global__ __launch_bounds__(THREADS_K1)
void tokenTypeLoss_k1(const float* __restrict__ outp, float4* __restrict__ partials) {
  __shared__ __align__(16) float smem[SMEM_FLOATS];

  const int chunk = blockIdx.x;
  const int b     = blockIdx.y;
  const int tid   = threadIdx.x;
  const int lane  = tid & 31;
  const int w     = __builtin_amdgcn_readfirstlane(tid >> 5);   // wave-uniform SGPR

  const int  c0 = chunk * ROWS_PER_CHUNK + w * ROWS_PER_WAVE;   // first row of this wave
  const uint64_t gbase =
      (uint64_t)(uintptr_t)outp + ((uint64_t)b * C_DIM + (uint64_t)c0) * (uint64_t)ROW_BYTES;
  // Low 32 bits of a generic pointer to LDS == LDS byte offset (aperture layout).
  const uint32_t lds_base = (uint32_t)(uintptr_t)smem + (uint32_t)w * WAVE_LDS_BYTES;

  const float NEG_INF = neg_inf_f();
  const bool active = (lane < 30);            // 30 lanes * 4 floats cover S=120

  // Per-lane column accumulators for s = 4*lane + i
  float cmax[4], csum[4], best[4];
  int   bidx[4];
#pragma unroll
  for (int i = 0; i < 4; ++i) { cmax[i] = NEG_INF; csum[i] = 0.0f; best[i] = NEG_INF; bidx[i] = 0; }

#if USE_TDM
  tdm_issue_tile(gbase, lds_base);            // prefetch tile 0 into buf 0
#endif

  for (int tile = 0; tile < NTILES; ++tile) {
#if USE_TDM
    if (tile + 1 < NTILES) {
      tdm_issue_tile(gbase + (uint64_t)(tile + 1) * TILE_BYTES,
                     lds_base + (uint32_t)(((tile + 1) & 1) * TILE_BYTES));
      __builtin_amdgcn_s_wait_tensorcnt(1);   // tile `tile` landed (TDM in-order per wave)
    } else {
      __builtin_amdgcn_s_wait_tensorcnt(0);
    }
    const float* tbuf = smem + (w * WAVE_LDS_BYTES + (tile & 1) * TILE_BYTES) / 4;
#endif
#pragma unroll
    for (int r = 0; r < TILE_ROWS; ++r) {
      const int c = c0 + tile * TILE_ROWS + r;
      float x[4] = {NEG_INF, NEG_INF, NEG_INF, NEG_INF};
      if (active) {
#if USE_TDM
        const float4 v = *(const float4*)(tbuf + r * S_DIM + lane * 4);
#else
        const float4 v = *(const float4*)(outp + ((size_t)b * C_DIM + (size_t)c) * S_DIM + lane * 4);
#endif
        x[0] = v.x; x[1] = v.y; x[2] = v.z; x[3] = v.w;
      }
      // Row logsumexp over S (wave allreduce; inactive lanes contribute -inf/0)
      float rmax = fmaxf(fmaxf(x[0], x[1]), fmaxf(x[2], x[3]));
#pragma unroll
      for (int off = 16; off > 0; off >>= 1) rmax = fmaxf(rmax, __shfl_xor(rmax, off, 32));
      float rsum = __expf(x[0] - rmax) + __expf(x[1] - rmax) +
                   __expf(x[2] - rmax) + __expf(x[3] - rmax);   // exp(-inf)=0 for idle lanes
#pragma unroll
      for (int off = 16; off > 0; off >>= 1) rsum += __shfl_xor(rsum, off, 32);
      const float lseS = rmax + __logf(rsum);

      if (active) {
#pragma unroll
        for (int i = 0; i < 4; ++i) {
          const float xv = x[i];
          // online logsumexp over C (renormalize only when max moves: ~1 exp/elem)
          if (xv > cmax[i]) { csum[i] = csum[i] * __expf(cmax[i] - xv) + 1.0f; cmax[i] = xv; }
          else              { csum[i] += __expf(xv - cmax[i]); }
          // argmax_c of (x - lse_S); strict '>' keeps first (lowest c) on ties
          const float sc = xv - lseS;
          if (sc > best[i]) { best[i] = sc; bidx[i] = c; }
        }
      }
    }
  }

  // Cross-wave merge through LDS (tiles no longer needed).
  __syncthreads();
  float4* stage = (float4*)smem;              // [WAVES_K1][128]
  if (active) {
#pragma unroll
    for (int i = 0; i < 4; ++i)
      stage[w * 128 + lane * 4 + i] = make_float4(cmax[i], csum[i], best[i], __int_as_float(bidx[i]));
  }
  __syncthreads();

  if (tid < S_DIM) {
    float m = NEG_INF, sum = 0.0f, bs = NEG_INF; int bc = 0;
    for (int ww = 0; ww < WAVES_K1; ++ww) {   // ascending wave order == ascending c
      const float4 p = stage[ww * 128 + tid];
      const float M = fmaxf(m, p.x);
      sum = sum * __expf(m - M) + p.y * __expf(p.x - M);
      m = M;
      if (p.z > bs) { bs = p.z; bc = __float_as_int(p.w); }
    }
    partials[(b * NCHUNK + chunk) * S_DIM + tid] = make_float4(m, sum, bs, __int_as_float(bc));
  }
}

// ---------------------------------------------------------------------------
// k2: per-b block merges NCHUNK partials per s, forms nll + type-mask, and
// block-reduces to per-b sums.
// ---------------------------------------------------------------------------
__global__ __launch_bounds__(128)
void tokenTypeLoss_k2(const float4* __restrict__ partials, const float* __restrict__ outp,
                      const int* __restrict__ target, const int* __restrict__ ttype,
                      float* __restrict__ bsums) {
  const int b = blockIdx.x;
  const int t = threadIdx.x;
  const float NEG_INF = neg_inf_f();

  float nll = 0.0f, msk = 0.0f;
  if (t < S_DIM) {
    float m = NEG_INF, sum = 0.0f, bs = NEG_INF; int bc = 0;
    for (int c = 0; c < NCHUNK; ++c) {        // ascending chunk order == ascending c
      const float4 p = partials[(b * NCHUNK + c) * S_DIM + t];
      const float M = fmaxf(m, p.x);
      sum = sum * __expf(m - M) + p.y * __expf(p.x - M);
      m = M;
      if (p.z > bs) { bs = p.z; bc = __float_as_int(p.w); }
    }
    const float lseC = m + __logf(sum);
    const int tg = target[b * S_DIM + t];     // harness delivers integer inputs as int32
    const float xt = outp[((size_t)b * C_DIM + (size_t)tg) * S_DIM + t];
    nll = lseC - xt;
    msk = (ttype[bc] != ttype[tg]) ? 1.0f : 0.0f;
  }

#pragma unroll
  for (int off = 16; off > 0; off >>= 1) {
    nll += __shfl_xor(nll, off, 32);
    msk += __shfl_xor(msk, off, 32);
  }
  __shared__ float sn[4], sm[4];
  const int wv = t >> 5, ln = t & 31;
  if (ln == 0) { sn[wv] = nll; sm[wv] = msk; }
  __syncthreads();
  if (t == 0) {
    bsums[b * 2 + 0] = sn[0] + sn[1] + sn[2] + sn[3];
    bsums[b * 2 + 1] = sm[0] + sm[1] + sm[2] + sm[3];
  }
}

// ---------------------------------------------------------------------------
// k3: one wave folds the 32 per-b sums into the scalar result.
// ---------------------------------------------------------------------------
__global__ __launch_bounds__(32)
void tokenTypeLoss_k3(const float* __restrict__ bsums, float* __restrict__ o) {
  const int t = threadIdx.x;
  float n = bsums[t * 2 + 0];
  float m = bsums[t * 2 + 1];
#pragma unroll
  for (int off = 16; off > 0; off >>= 1) {
    n += __shfl_xor(n, off, 32);
    m += __shfl_xor(m, off, 32);
  }
  if (t == 0) {
    const float loss = n * (1.0f / (float)BS_TOTAL);
    const float mm   = m * (1.0f / (float)BS_TOTAL);
    o[0] = loss * (1.0f + mm);                // loss + WEIGHT*loss*mask_mean, WEIGHT=1
  }
}

extern "C" void kernel_launch(void* const* d_in, const int* in_sizes, int n_in,
                              void* d_out, int out_size, void* d_ws, size_t ws_size,
                              hipStream_t stream) {
  (void)in_sizes; (void)n_in; (void)out_size; (void)ws_size;
  const float* outp   = (const float*)d_in[0];   // [32, 8192, 120] f32
  const int*   target = (const int*)d_in[1];     // [32, 120] int
  const int*   ttype  = (const int*)d_in[2];     // [8192] int

  float4* partials = (float4*)d_ws;              // 32*16*120 float4 = 0.94 MB
  float*  bsums    = (float*)((char*)d_ws + (size_t)B_DIM * NCHUNK * S_DIM * sizeof(float4));

  dim3 g1(NCHUNK, B_DIM);
  tokenTypeLoss_k1<<<g1, THREADS_K1, 0, stream>>>(outp, partials);
  tokenTypeLoss_k2<<<B_DIM, 128, 0, stream>>>(partials, outp, target, ttype, bsums);
  tokenTypeLoss_k3<<<1, 32, 0, stream>>>(bsums, (float*)d_out);
}